// RotationAttention_44839458570374
// MI455X (gfx1250) — compile-verified
//
#include <hip/hip_runtime.h>
#include <math.h>

#define M_ROWS 384
#define DIM    512
#define INV_T  10.0f

typedef __attribute__((ext_vector_type(2))) float v2f;
typedef __attribute__((ext_vector_type(8))) float v8f;

// ---------------------------------------------------------------------------
// K1: Q = img - F  (per-row), exact 3x3 inverse of refer_r via adjugate.
// ---------------------------------------------------------------------------
__global__ void rotattn_prep(const float* __restrict__ img,
                             const float* __restrict__ F,
                             const float* __restrict__ Rr,
                             float* __restrict__ Q,
                             float* __restrict__ Rinv) {
    const int a = blockIdx.x;
    const int tid = threadIdx.x;
    for (int d = tid; d < DIM; d += blockDim.x)
        Q[a * DIM + d] = img[d] - F[a * DIM + d];
    if (tid == 0) {
        const float* R = Rr + a * 9;
        float c00 =   R[4]*R[8] - R[5]*R[7];
        float c01 = -(R[3]*R[8] - R[5]*R[6]);
        float c02 =   R[3]*R[7] - R[4]*R[6];
        float det = R[0]*c00 + R[1]*c01 + R[2]*c02;
        float id  = 1.0f / det;
        float* O = Rinv + a * 9;
        O[0] =  c00 * id;
        O[3] =  c01 * id;
        O[6] =  c02 * id;
        O[1] = -(R[1]*R[8] - R[2]*R[7]) * id;
        O[4] =  (R[0]*R[8] - R[2]*R[6]) * id;
        O[7] = -(R[0]*R[7] - R[1]*R[6]) * id;
        O[2] =  (R[1]*R[5] - R[2]*R[4]) * id;
        O[5] = -(R[0]*R[5] - R[2]*R[3]) * id;
        O[8] =  (R[0]*R[4] - R[1]*R[3]) * id;
    }
}

// ---------------------------------------------------------------------------
// K2: S = (Q @ F^T) / TEMPERATURE   (384x384, K=512), f32 WMMA.
// One wave per 16x16 tile. A = Q rows (row-major), B = F^T so B[k][n]=F[n][k]
// -> both fragments load contiguous float2 pairs.
// A frag: lane(0-15)->M=lane,K={k,k+1}; lane(16-31)->M=lane-16,K={k+2,k+3}.
// C/D: vgpr i -> M = i + 8*(lane>=16), N = lane&15.
// ---------------------------------------------------------------------------
__global__ void rotattn_s_gemm(const float* __restrict__ Q,
                               const float* __restrict__ F,
                               float* __restrict__ S) {
    const int wid  = (blockIdx.x * blockDim.x + threadIdx.x) >> 5; // 576 waves
    const int lane = threadIdx.x & 31;
    const int tm = (wid / 24) * 16;
    const int tn = (wid % 24) * 16;
    const int l15   = lane & 15;
    const int khalf = (lane >> 4) * 2;

    const float* qrow = Q + (tm + l15) * DIM + khalf;
    const float* frow = F + (tn + l15) * DIM + khalf;

    v8f acc = {};
#pragma unroll 8
    for (int k = 0; k < DIM; k += 4) {
        v2f a = *(const v2f*)(qrow + k);
        v2f b = *(const v2f*)(frow + k);
        acc = __builtin_amdgcn_wmma_f32_16x16x4_f32(
            false, a, false, b, (short)0, acc, false, false);
    }
    const int r0 = tm + (lane >> 4) * 8;
    const int c  = tn + l15;
#pragma unroll
    for (int i = 0; i < 8; ++i)
        S[(r0 + i) * M_ROWS + c] = acc[i] * INV_T;
}

// ---------------------------------------------------------------------------
// K3: per-row factorized softmax + bilinear collapse + We projection.
// p = softmax(S[a,:]), q = softmax(-S[a,:]);
// Pbar = sum_b p[b]*R[b], Qbar = sum_b q[b]*Rinv[b];
// v6 = (Pbar @ Qbar) rows 0..1; OUT[a,:] = v6 @ We + be.
// One 256-thread block per row a.
// ---------------------------------------------------------------------------
__global__ void rotattn_softmax_proj(const float* __restrict__ S,
                                     const float* __restrict__ Rr,
                                     const float* __restrict__ Rinv,
                                     const float* __restrict__ We,
                                     const float* __restrict__ be,
                                     float* __restrict__ OUT) {
    const int a = blockIdx.x;
    const int tid = threadIdx.x;
    __shared__ float red[256];
    __shared__ float v6s[6];

    const float* srow = S + a * M_ROWS;

    float lmax = -3.0e38f, lmin = 3.0e38f;
    for (int b = tid; b < M_ROWS; b += 256) {
        float v = srow[b];
        lmax = fmaxf(lmax, v);
        lmin = fminf(lmin, v);
    }
    red[tid] = lmax; __syncthreads();
    for (int s = 128; s > 0; s >>= 1) {
        if (tid < s) red[tid] = fmaxf(red[tid], red[tid + s]);
        __syncthreads();
    }
    const float rmax = red[0]; __syncthreads();
    red[tid] = lmin; __syncthreads();
    for (int s = 128; s > 0; s >>= 1) {
        if (tid < s) red[tid] = fminf(red[tid], red[tid + s]);
        __syncthreads();
    }
    const float rmin = red[0]; __syncthreads();

    float sp = 0.0f, sq = 0.0f;
    float accP[9] = {0,0,0,0,0,0,0,0,0};
    float accQ[9] = {0,0,0,0,0,0,0,0,0};
    for (int b = tid; b < M_ROWS; b += 256) {
        const float v = srow[b];
        const float p = expf(v - rmax);
        const float q = expf(rmin - v);
        sp += p; sq += q;
        const float* rb = Rr   + b * 9;
        const float* ib = Rinv + b * 9;
#pragma unroll
        for (int j = 0; j < 9; ++j) {
            accP[j] += p * rb[j];
            accQ[j] += q * ib[j];
        }
    }
    // reduce 20 scalars across the block
    float vals[20];
    vals[0] = sp; vals[1] = sq;
#pragma unroll
    for (int j = 0; j < 9; ++j) { vals[2 + j] = accP[j]; vals[11 + j] = accQ[j]; }
#pragma unroll 1
    for (int v = 0; v < 20; ++v) {
        red[tid] = vals[v]; __syncthreads();
        for (int s = 128; s > 0; s >>= 1) {
            if (tid < s) red[tid] += red[tid + s];
            __syncthreads();
        }
        vals[v] = red[0]; __syncthreads();
    }

    if (tid == 0) {
        const float isp = 1.0f / vals[0];
        const float isq = 1.0f / vals[1];
        float Pb[9], Qb[9];
#pragma unroll
        for (int j = 0; j < 9; ++j) { Pb[j] = vals[2 + j] * isp; Qb[j] = vals[11 + j] * isq; }
#pragma unroll
        for (int i = 0; i < 2; ++i)
#pragma unroll
            for (int k = 0; k < 3; ++k)
                v6s[i * 3 + k] = Pb[i*3+0]*Qb[0*3+k] + Pb[i*3+1]*Qb[1*3+k] + Pb[i*3+2]*Qb[2*3+k];
    }
    __syncthreads();

    const float t0 = v6s[0], t1 = v6s[1], t2 = v6s[2],
                t3 = v6s[3], t4 = v6s[4], t5 = v6s[5];
    for (int d = tid; d < DIM; d += 256) {
        float o = be[d];
        o += t0 * We[0 * DIM + d];
        o += t1 * We[1 * DIM + d];
        o += t2 * We[2 * DIM + d];
        o += t3 * We[3 * DIM + d];
        o += t4 * We[4 * DIM + d];
        o += t5 * We[5 * DIM + d];
        OUT[a * DIM + d] = o;
    }
}

// ---------------------------------------------------------------------------
// K4: H = relu(OUT @ W1 + b1)   (384x512, K=512), f32 WMMA.
// B = W1 row-major: B[k][n] = W1[k*512+n] (column-strided fragment loads).
// ---------------------------------------------------------------------------
__global__ void rotattn_mlp1(const float* __restrict__ X,
                             const float* __restrict__ W1,
                             const float* __restrict__ b1,
                             float* __restrict__ H) {
    const int wid  = (blockIdx.x * blockDim.x + threadIdx.x) >> 5; // 768 waves
    const int lane = threadIdx.x & 31;
    const int tm = (wid >> 5) * 16;        // 24 tiles in M
    const int tn = (wid & 31) * 16;        // 32 tiles in N
    const int l15   = lane & 15;
    const int khalf = (lane >> 4) * 2;

    const float* xrow = X + (tm + l15) * DIM + khalf;
    const float* wcol = W1 + (size_t)khalf * DIM + (tn + l15);

    v8f acc = {};
#pragma unroll 4
    for (int k = 0; k < DIM; k += 4) {
        v2f a = *(const v2f*)(xrow + k);
        v2f b;
        b.x = wcol[(size_t)k * DIM];
        b.y = wcol[(size_t)k * DIM + DIM];
        acc = __builtin_amdgcn_wmma_f32_16x16x4_f32(
            false, a, false, b, (short)0, acc, false, false);
    }
    const int r0 = tm + (lane >> 4) * 8;
    const int c  = tn + l15;
    const float bias = b1[c];
#pragma unroll
    for (int i = 0; i < 8; ++i)
        H[(r0 + i) * DIM + c] = fmaxf(acc[i] + bias, 0.0f);
}

// ---------------------------------------------------------------------------
// K5: r6 = H @ W2 + b2 ; recover_R (reference's exact un-normalized
// projection order) ; final R @ Rr -> d_out. One block per row.
// ---------------------------------------------------------------------------
__global__ void rotattn_head(const float* __restrict__ H,
                             const float* __restrict__ W2,
                             const float* __restrict__ b2,
                             const float* __restrict__ Rr,
                             float* __restrict__ out) {
    const int a = blockIdx.x;
    const int tid = threadIdx.x;
    __shared__ float red[256];
    float part[6] = {0,0,0,0,0,0};
    for (int k = tid; k < DIM; k += 256) {
        const float hv = H[a * DIM + k];
#pragma unroll
        for (int t = 0; t < 6; ++t) part[t] += hv * W2[k * 6 + t];
    }
    float r6[6];
#pragma unroll 1
    for (int t = 0; t < 6; ++t) {
        red[tid] = part[t]; __syncthreads();
        for (int s = 128; s > 0; s >>= 1) {
            if (tid < s) red[tid] += red[tid + s];
            __syncthreads();
        }
        if (tid == 0) r6[t] = red[0] + b2[t];
        __syncthreads();
    }
    if (tid == 0) {
        float r1[3] = { r6[0], r6[1], r6[2] };
        float r2[3] = { r6[3], r6[4], r6[5] };
        const float d = r1[0]*r2[0] + r1[1]*r2[1] + r1[2]*r2[2];
        r2[0] -= d * r1[0]; r2[1] -= d * r1[1]; r2[2] -= d * r1[2];
        float r3[3] = { r1[1]*r2[2] - r1[2]*r2[1],
                        r1[2]*r2[0] - r1[0]*r2[2],
                        r1[0]*r2[1] - r1[1]*r2[0] };
        float R[9];
        const float n1 = 1.0f / fmaxf(sqrtf(r1[0]*r1[0]+r1[1]*r1[1]+r1[2]*r1[2]), 1e-12f);
        const float n2 = 1.0f / fmaxf(sqrtf(r2[0]*r2[0]+r2[1]*r2[1]+r2[2]*r2[2]), 1e-12f);
        const float n3 = 1.0f / fmaxf(sqrtf(r3[0]*r3[0]+r3[1]*r3[1]+r3[2]*r3[2]), 1e-12f);
        R[0]=r1[0]*n1; R[1]=r1[1]*n1; R[2]=r1[2]*n1;
        R[3]=r2[0]*n2; R[4]=r2[1]*n2; R[5]=r2[2]*n2;
        R[6]=r3[0]*n3; R[7]=r3[1]*n3; R[8]=r3[2]*n3;
        const float* A = Rr + a * 9;
#pragma unroll
        for (int i = 0; i < 3; ++i)
#pragma unroll
            for (int k = 0; k < 3; ++k)
                out[a * 9 + i * 3 + k] =
                    R[i*3+0]*A[0*3+k] + R[i*3+1]*A[1*3+k] + R[i*3+2]*A[2*3+k];
    }
}

// ---------------------------------------------------------------------------
extern "C" void kernel_launch(void* const* d_in, const int* in_sizes, int n_in,
                              void* d_out, int out_size, void* d_ws, size_t ws_size,
                              hipStream_t stream) {
    const float* img = (const float*)d_in[0];   // (1, 512)
    const float* F   = (const float*)d_in[1];   // (384, 512)
    const float* Rr  = (const float*)d_in[2];   // (384, 9)
    const float* We  = (const float*)d_in[3];   // (6, 512)
    const float* be  = (const float*)d_in[4];   // (512,)
    const float* W1  = (const float*)d_in[5];   // (512, 512)
    const float* b1  = (const float*)d_in[6];   // (512,)
    const float* W2  = (const float*)d_in[7];   // (512, 6)
    const float* b2  = (const float*)d_in[8];   // (6,)

    float* ws   = (float*)d_ws;
    float* Rinv = ws;                               // 3456 floats (pad to 4096)
    float* Q    = ws + 4096;                        // 196608
    float* S    = Q  + (size_t)M_ROWS * DIM;        // 147456
    float* OUT  = S  + (size_t)M_ROWS * M_ROWS;     // 196608
    float* H    = OUT + (size_t)M_ROWS * DIM;       // 196608

    rotattn_prep<<<M_ROWS, 256, 0, stream>>>(img, F, Rr, Q, Rinv);
    // 24x24 tiles -> 576 waves -> 72 blocks of 8 waves
    rotattn_s_gemm<<<72, 256, 0, stream>>>(Q, F, S);
    rotattn_softmax_proj<<<M_ROWS, 256, 0, stream>>>(S, Rr, Rinv, We, be, OUT);
    // 24x32 tiles -> 768 waves -> 96 blocks of 8 waves
    rotattn_mlp1<<<96, 256, 0, stream>>>(OUT, W1, b1, H);
    rotattn_head<<<M_ROWS, 256, 0, stream>>>(H, W2, b2, Rr, (float*)d_out);
}